// ViT_30554397344360
// MI455X (gfx1250) — compile-verified
//
#include <hip/hip_runtime.h>
#include <hip/hip_bf16.h>
#include <math.h>

// ---------------- problem constants ----------------
#define N_IMG   128
#define CCH     3
#define IMGHW   255
#define PATCH   5
#define NP      51          // patches per side
#define SPATCH  (NP*NP)     // 2601
#define SEQ     (SPATCH+1)  // 2602
#define HID     128
#define KDIM    75          // 3*5*5
#define KSTEPS  19          // ceil(75/4) k-steps of 4 (k=75 zero-padded)
#define LDK     84          // LDS row stride in floats (bank-conflict free)
#define MTILE   16
#define NTILES  ((SPATCH + MTILE - 1) / MTILE)   // 163
#define CHUNKS  21
#define TPC     8           // tiles per chunk: 21*8 = 168 >= 163
#define STAGE_N (MTILE*KDIM)             // 1200 elements staged per tile
#define STAGE_U ((STAGE_N + 255) / 256)  // 5 staging slots per thread

typedef __attribute__((ext_vector_type(2))) float v2f;
typedef __attribute__((ext_vector_type(8))) float v8f;

// D = A(16x4) * B(4x16) + C, all f32, wave32.
// A frag: lane L holds row L%16, K = kk + (L<16 ? 0 : 2) and +1  -> ds_load_b64
// B frag: lane L holds col L%16 (row of W), same K selection      -> held in VGPRs
// C/D:    VGPR r, lane L -> M = r + (L<16 ? 0 : 8), N = L%16

__global__ __launch_bounds__(256)
void vit_embed_kernel(const float* __restrict__ img,
                      const float* __restrict__ W,
                      const float* __restrict__ bias,
                      float* __restrict__ out) {
    __shared__ float sW[HID * LDK];    // 43,008 B (only used during init)
    __shared__ float sP[MTILE * LDK];  //  5,376 B

    const int tid   = threadIdx.x;
    const int chunk = blockIdx.x;
    const int n     = blockIdx.y;

    // Zero LDS once: guarantees k = 75..(LDK-1) pad columns are 0 forever.
    for (int i = tid; i < HID * LDK; i += 256)   sW[i] = 0.0f;
    for (int i = tid; i < MTILE * LDK; i += 256) sP[i] = 0.0f;
    __syncthreads();

    // Stage W[128,75] row-major into LDS with stride LDK.
    for (int idx = tid; idx < HID * KDIM; idx += 256) {
        int h = idx / KDIM;
        int k = idx - h * KDIM;
        sW[h * LDK + k] = W[idx];
    }
    __syncthreads();

    const int wave = tid >> 5;          // 0..7 -> N-tile (hidden block)
    const int lane = tid & 31;
    const int hcol = (wave << 4) + (lane & 15);          // hidden index 0..127
    const float bb = bias[hcol];
    // pos freq: 10000^(-(h & ~1)/128);  phase pi/2 turns sin into cos for odd h
    const float fr    = __expf(-9.210340371976184f * (float)(hcol & ~1) * (1.0f / 128.0f));
    const float phase = (hcol & 1) ? 1.5707963267948966f : 0.0f;

    const float* __restrict__ imgN = img + (size_t)n * (CCH * IMGHW * IMGHW);

    const int asel = (lane < 16) ? 0 : 2;
    const float* __restrict__ ap0 = &sP[(lane & 15) * LDK + asel];

    // ---- hoist all 19 B fragments (W) into registers: tile-invariant ----
    v2f Bfrag[KSTEPS];
    {
        const float* __restrict__ bp = &sW[hcol * LDK + asel];
#pragma unroll
        for (int s = 0; s < KSTEPS; ++s)
            Bfrag[s] = *(const v2f*)(bp + 4 * s);
    }

    // ---- tile-invariant im2col decomposition for this thread's staging slots ----
    int  slArr[STAGE_U];    // patch row inside tile
    int  lofs[STAGE_U];     // LDS offset
    int  gofs[STAGE_U];     // (c*255 + i)*255 + j  part of the global offset
    bool valid[STAGE_U];
#pragma unroll
    for (int u = 0; u < STAGE_U; ++u) {
        int idx  = tid + u * 256;
        valid[u] = (idx < STAGE_N);
        int id2  = valid[u] ? idx : 0;
        int sl   = id2 / KDIM;
        int k    = id2 - sl * KDIM;
        int c    = k / 25;
        int r    = k - c * 25;
        int i    = r / 5;
        int j    = r - i * 5;
        slArr[u] = sl;
        lofs[u]  = sl * LDK + k;
        gofs[u]  = (c * IMGHW + i) * IMGHW + j;
    }

    // software pipeline: global loads for tile t+1 overlap compute of tile t
    float stage[STAGE_U];
    {
        const int mtile0 = chunk * TPC;
#pragma unroll
        for (int u = 0; u < STAGE_U; ++u) {
            if (valid[u]) {
                int m = mtile0 * MTILE + slArr[u];
                if (m > SPATCH - 1) m = SPATCH - 1;
                int py = m / NP;
                int px = m - py * NP;
                stage[u] = __builtin_nontemporal_load(
                    imgN + gofs[u] + py * (PATCH * IMGHW) + px * PATCH);
            }
        }
    }

    for (int t = 0; t < TPC; ++t) {
        const int mtile = chunk * TPC + t;
        if (mtile * MTILE >= SPATCH) break;   // uniform per block

        // ---- commit staged registers to LDS ----
        __syncthreads();   // previous tile's LDS readers done
#pragma unroll
        for (int u = 0; u < STAGE_U; ++u)
            if (valid[u]) sP[lofs[u]] = stage[u];
        __syncthreads();

        // ---- issue next tile's global loads (latency hidden under compute) ----
        const int mnext = mtile + 1;
        if (t + 1 < TPC && mnext * MTILE < SPATCH) {
#pragma unroll
            for (int u = 0; u < STAGE_U; ++u) {
                if (valid[u]) {
                    int m = mnext * MTILE + slArr[u];
                    if (m > SPATCH - 1) m = SPATCH - 1;
                    int py = m / NP;
                    int px = m - py * NP;
                    stage[u] = __builtin_nontemporal_load(
                        imgN + gofs[u] + py * (PATCH * IMGHW) + px * PATCH);
                }
            }
        }

        // ---- 19 k-steps of v_wmma_f32_16x16x4_f32, 2 accumulator chains ----
        v8f acc0 = {};
        v8f acc1 = {};
#pragma unroll
        for (int s = 0; s < KSTEPS; ++s) {
            v2f a = *(const v2f*)(ap0 + 4 * s);
            if (s & 1)
                acc1 = __builtin_amdgcn_wmma_f32_16x16x4_f32(
                           false, a, false, Bfrag[s], (short)0, acc1, false, false);
            else
                acc0 = __builtin_amdgcn_wmma_f32_16x16x4_f32(
                           false, a, false, Bfrag[s], (short)0, acc0, false, false);
        }

        // ---- epilogue: + bias + positional embedding (Cody-Waite + v_sin), store ----
        const int mbase = mtile * MTILE + ((lane < 16) ? 0 : 8);
        float* __restrict__ orow =
            out + ((size_t)n * SEQ + (size_t)(mbase + 1)) * HID + hcol;

        const float INV2PI = 0.15915494309189535f;
        const float C1     = 6.28125f;          // 2*pi split: exact 9-bit head
        const float C2     = 1.9353072e-3f;     // tail

        if (mtile < NTILES - 1) {
#pragma unroll
            for (int r = 0; r < 8; ++r) {
                float ang = fmaf((float)(mbase + r + 1), fr, phase);
                float q   = rintf(ang * INV2PI);
                float rd  = fmaf(-C1, q, ang);
                rd        = fmaf(-C2, q, rd);
                float pe  = __sinf(rd);
                __builtin_nontemporal_store(acc0[r] + acc1[r] + bb + pe,
                                            orow + (size_t)r * HID);
            }
        } else {
#pragma unroll
            for (int r = 0; r < 8; ++r) {
                int m = mbase + r;
                if (m < SPATCH) {
                    float ang = fmaf((float)(m + 1), fr, phase);
                    float q   = rintf(ang * INV2PI);
                    float rd  = fmaf(-C1, q, ang);
                    rd        = fmaf(-C2, q, rd);
                    float pe  = __sinf(rd);
                    __builtin_nontemporal_store(acc0[r] + acc1[r] + bb + pe,
                                                orow + (size_t)r * HID);
                }
            }
        }
    }
}

// Row 0 of every image: class_token + pos[0,:] (= 0 for even h, 1 for odd h)
__global__ __launch_bounds__(256)
void vit_cls_kernel(const float* __restrict__ ct, float* __restrict__ out) {
    int g = blockIdx.x * blockDim.x + threadIdx.x;   // 0 .. 128*128-1
    if (g < N_IMG * HID) {
        int n = g >> 7;
        int h = g & 127;
        out[(size_t)n * SEQ * HID + h] = ct[h] + ((h & 1) ? 1.0f : 0.0f);
    }
}

extern "C" void kernel_launch(void* const* d_in, const int* in_sizes, int n_in,
                              void* d_out, int out_size, void* d_ws, size_t ws_size,
                              hipStream_t stream) {
    const float* images = (const float*)d_in[0];   // [128,3,255,255]
    const float* W      = (const float*)d_in[1];   // [128,75]
    const float* b      = (const float*)d_in[2];   // [128]
    const float* ct     = (const float*)d_in[3];   // [1,128]
    float* out          = (float*)d_out;           // [128,2602,128]

    dim3 grid(CHUNKS, N_IMG);
    vit_embed_kernel<<<grid, 256, 0, stream>>>(images, W, b, out);

    int tot = N_IMG * HID;
    vit_cls_kernel<<<(tot + 255) / 256, 256, 0, stream>>>(ct, out);
}